// STCGNN_16647293239401
// MI455X (gfx1250) — compile-verified
//
#include <hip/hip_runtime.h>
#include <math.h>

// ---------------- problem constants (from reference) ----------------
#define B_    8
#define T_    12
#define N_    80
#define C_    12
#define HID_  32
#define OUTH_ 6
#define ALPHA_ 3.0f
#define BT_   (B_*T_)        // 96
#define ROWS_ (B_*N_*C_)     // 7680
#define BN_   (B_*N_)        // 640
#define N2_   (N_*N_)        // 6400
#define C2_   (C_*C_)        // 144
#define ICH_  40             // split-K chunks for the 6400x6400 fusion matvec

typedef __attribute__((ext_vector_type(2))) float v2f;
typedef __attribute__((ext_vector_type(8))) float v8f;

// fp32 WMMA: D(16x16) += A(16x4) * B(4x16), wave32, fp32 end-to-end.
static __device__ __forceinline__ v8f wmma4(v2f a, v2f b, v8f c) {
    return __builtin_amdgcn_wmma_f32_16x16x4_f32(false, a, false, b, (short)0, c,
                                                 false, false);
}
static __device__ __forceinline__ float sigf(float x) { return 1.0f / (1.0f + expf(-x)); }
static __device__ __forceinline__ v2f ld2(const float* p) { return *(const v2f*)p; }

// ---------------- phase A: graph learning ----------------

// Us/Vs = tanh(3 * X @ W) over channel dim (12-dot), layout [bt][n][h] (80 rows exact)
__global__ void k_uv_spatial(const float* __restrict__ X, const float* __restrict__ Wu,
                             const float* __restrict__ Wv, float* __restrict__ U,
                             float* __restrict__ V) {
    int idx = blockIdx.x * blockDim.x + threadIdx.x;
    if (idx >= BT_ * N_ * HID_) return;
    int h  = idx & (HID_ - 1);
    int n  = (idx >> 5) % N_;
    int bt = idx / (N_ * HID_);
    const float* x = X + (size_t)(bt * N_ + n) * C_;
    float su = 0.f, sv = 0.f;
    for (int c = 0; c < C_; ++c) {
        float xv = x[c];
        su += xv * Wu[c * HID_ + h];
        sv += xv * Wv[c * HID_ + h];
    }
    U[idx] = tanhf(ALPHA_ * su);
    V[idx] = tanhf(ALPHA_ * sv);
}

// Uc/Vc padded to 16 rows per slice (rows 12..15 zero) -> guard-free WMMA loads later
__global__ void k_uv_channel(const float* __restrict__ X, const float* __restrict__ Wu,
                             const float* __restrict__ Wv, float* __restrict__ U,
                             float* __restrict__ V) {
    int idx = blockIdx.x * blockDim.x + threadIdx.x;
    if (idx >= BT_ * 16 * HID_) return;
    int h  = idx & (HID_ - 1);
    int c  = (idx >> 5) & 15;
    int bt = idx / (16 * HID_);
    float su = 0.f, sv = 0.f;
    if (c < C_) {
        for (int n = 0; n < N_; ++n) {
            float xv = X[(size_t)(bt * N_ + n) * C_ + c];
            su += xv * Wu[n * HID_ + h];
            sv += xv * Wv[n * HID_ + h];
        }
        U[idx] = tanhf(ALPHA_ * su);
        V[idx] = tanhf(ALPHA_ * sv);
    } else {
        U[idx] = 0.f;
        V[idx] = 0.f;
    }
}

// M[n,m] = sum_{bt,h} U[bt,n,h] * V[bt,m,h]; DIMP = padded per-slice rows (guard-free loads)
template<int DIM, int DIMP>
__global__ __launch_bounds__(32) void k_mm_wmma(const float* __restrict__ U,
                                                const float* __restrict__ V,
                                                float* __restrict__ M) {
    int lane = threadIdx.x, r = lane & 15, half = lane >> 4;
    int rtile = blockIdx.y, ctile = blockIdx.x;
    int arow = rtile * 16 + r;
    int bcol = ctile * 16 + r;
    v8f acc = {0.f, 0.f, 0.f, 0.f, 0.f, 0.f, 0.f, 0.f};
    for (int bt = 0; bt < BT_; ++bt) {
        const float* Ub = U + (size_t)bt * DIMP * HID_;
        const float* Vb = V + (size_t)bt * DIMP * HID_;
        #pragma unroll
        for (int k4 = 0; k4 < HID_; k4 += 4) {
            int k = k4 + 2 * half;
            v2f a = ld2(Ub + arow * HID_ + k);
            v2f b = ld2(Vb + bcol * HID_ + k);
            acc = wmma4(a, b, acc);
        }
    }
    for (int i = 0; i < 8; ++i) {
        int row = rtile * 16 + i + 8 * half;
        int col = ctile * 16 + r;
        if (row < DIM && col < DIM) M[(size_t)row * DIM + col] = acc[i];
    }
}

// P = softmax(relu(M - M^T)) row-wise
__global__ void k_softmax(const float* __restrict__ M, float* __restrict__ P, int dim) {
    __shared__ float red[128];
    int row = blockIdx.x, t = threadIdx.x;
    float mx = -1e30f;
    for (int i = t; i < dim; i += blockDim.x) {
        float d = M[(size_t)row * dim + i] - M[(size_t)i * dim + row];
        d = d > 0.f ? d : 0.f;
        mx = fmaxf(mx, d);
    }
    red[t] = mx; __syncthreads();
    for (int s = 64; s > 0; s >>= 1) { if (t < s) red[t] = fmaxf(red[t], red[t + s]); __syncthreads(); }
    mx = red[0]; __syncthreads();
    float sum = 0.f;
    for (int i = t; i < dim; i += blockDim.x) {
        float d = M[(size_t)row * dim + i] - M[(size_t)i * dim + row];
        d = d > 0.f ? d : 0.f;
        sum += expf(d - mx);
    }
    red[t] = sum; __syncthreads();
    for (int s = 64; s > 0; s >>= 1) { if (t < s) red[t] += red[t + s]; __syncthreads(); }
    sum = red[0]; __syncthreads();
    for (int i = t; i < dim; i += blockDim.x) {
        float d = M[(size_t)row * dim + i] - M[(size_t)i * dim + row];
        d = d > 0.f ? d : 0.f;
        P[(size_t)row * dim + i] = expf(d - mx) / sum;
    }
}

// Split-K partial for a_j = sigmoid(A.flat @ WA + bA + P.flat @ WP + bP)
// HBM-bound (328 MB spatial pair); coalesced over j, deterministic (no atomics).
__global__ void k_fuse_partial(const float* __restrict__ A,  const float* __restrict__ WA,
                               const float* __restrict__ P,  const float* __restrict__ WP,
                               float* __restrict__ PART, int n2, int rowsPerChunk) {
    int j  = blockIdx.x * 256 + threadIdx.x;
    int i0 = blockIdx.y * rowsPerChunk;
    int i1 = i0 + rowsPerChunk; if (i1 > n2) i1 = n2;
    float s = 0.f;
    if (j < n2) {
        for (int i = i0; i < i1; ++i)
            s += A[i] * WA[(size_t)i * n2 + j] + P[i] * WP[(size_t)i * n2 + j];
        PART[(size_t)blockIdx.y * n2 + j] = s;
    }
}

__global__ void k_fuse_final(const float* __restrict__ PART, int nchunks,
                             const float* __restrict__ bA, const float* __restrict__ bP,
                             const float* __restrict__ A,  const float* __restrict__ P,
                             float* __restrict__ G, int n2) {
    int j = blockIdx.x * 256 + threadIdx.x;
    if (j >= n2) return;
    float s = bA[j] + bP[j];
    for (int c = 0; c < nchunks; ++c) s += PART[(size_t)c * n2 + j];
    float a = sigf(s);
    G[j] = a * A[j] + (1.f - a) * P[j];
}

// Gs2 = 2 * Gs @ Gs - I   (80x80, WMMA, guard-free)
__global__ __launch_bounds__(32) void k_cheby(const float* __restrict__ G,
                                              float* __restrict__ G2) {
    int lane = threadIdx.x, r = lane & 15, half = lane >> 4;
    int rtile = blockIdx.y, ctile = blockIdx.x;
    int arow = rtile * 16 + r;
    int col  = ctile * 16 + r;
    v8f acc = {0.f, 0.f, 0.f, 0.f, 0.f, 0.f, 0.f, 0.f};
    #pragma unroll
    for (int k4 = 0; k4 < N_; k4 += 4) {
        int k = k4 + 2 * half;
        v2f a = ld2(G + arow * N_ + k);
        v2f b;
        b.x = G[k * N_ + col];
        b.y = G[(k + 1) * N_ + col];
        acc = wmma4(a, b, acc);
    }
    for (int i = 0; i < 8; ++i) {
        int row = rtile * 16 + i + 8 * half;
        G2[row * N_ + col] = 2.f * acc[i] - (row == col ? 1.f : 0.f);
    }
}

__global__ void k_fill0(float* __restrict__ p, int n) {
    int i = blockIdx.x * blockDim.x + threadIdx.x;
    if (i < n) p[i] = 0.f;
}

// Wp[k][n] = (k < Kact) ? W[k][n] : 0   (pads weights so GEMM loads are guard-free)
__global__ void k_padw(const float* __restrict__ W, int Kact, int Nw,
                       float* __restrict__ Wp, int Kpad) {
    int idx = blockIdx.x * blockDim.x + threadIdx.x;
    if (idx >= Kpad * Nw) return;
    int k = idx / Nw;
    Wp[idx] = (k < Kact) ? W[idx] : 0.f;
}

// ---------------- phase B: GRU cells ----------------
// XH layout: [b][n][CLP], entries cl = c*D + l for cl < C*D, zero-padded to CLP.

// encoder: D=33, CLP=400; l==0 -> X_seq[b,t,n,c], l>0 -> H
__global__ void k_xh_enc(const float* __restrict__ Xseq, int t, const float* __restrict__ H,
                         float* __restrict__ XH) {
    int idx = blockIdx.x * blockDim.x + threadIdx.x;
    if (idx >= BN_ * 400) return;
    int cl = idx % 400, bn = idx / 400;
    float v = 0.f;
    if (cl < 396) {
        int c = cl / 33, l = cl % 33;
        if (l == 0) {
            int b = bn / N_, n = bn % N_;
            v = Xseq[(size_t)((b * T_ + t) * N_ + n) * C_ + c];
        } else {
            v = H[(size_t)(bn * C_ + c) * HID_ + (l - 1)];
        }
    }
    XH[idx] = v;
}

// decoder: Xin == H, XH = [H | H], D=64, CLP=768 (no padding)
__global__ void k_xh_dec(const float* __restrict__ H, float* __restrict__ XH) {
    int idx = blockIdx.x * blockDim.x + threadIdx.x;
    if (idx >= BN_ * 768) return;
    int cl = idx % 768, bn = idx / 768;
    int c = cl >> 6, l = cl & 63;
    XH[idx] = H[(size_t)(bn * C_ + c) * HID_ + (l & 31)];
}

// Xn[g][b][m][clp] = sum_n G_g[n,m] * XH[b][n][clp]; guard-free WMMA
template<int CLP>
__global__ __launch_bounds__(32) void k_diffuse(const float* __restrict__ XH,
                                                const float* __restrict__ Gs,
                                                const float* __restrict__ Gs2,
                                                float* __restrict__ XN) {
    int lane = threadIdx.x, r = lane & 15, half = lane >> 4;
    int ctile = blockIdx.x, mtile = blockIdx.y;
    int bz = blockIdx.z, b = bz >> 1, g = bz & 1;
    const float* G = g ? Gs2 : Gs;
    const float* X = XH + (size_t)b * N_ * CLP;
    float* Y = XN + ((size_t)g * B_ + b) * N_ * CLP;
    int m   = mtile * 16 + r;
    int col = ctile * 16 + r;
    v8f acc = {0.f, 0.f, 0.f, 0.f, 0.f, 0.f, 0.f, 0.f};
    #pragma unroll 5
    for (int k4 = 0; k4 < N_; k4 += 4) {
        int k = k4 + 2 * half;
        v2f a, bf;
        a.x = G[k * N_ + m];               // A[m,n] = G[n,m]
        a.y = G[(k + 1) * N_ + m];
        bf.x = X[(size_t)k * CLP + col];
        bf.y = X[(size_t)(k + 1) * CLP + col];
        acc = wmma4(a, bf, acc);
    }
    for (int i = 0; i < 8; ++i) {
        int row = mtile * 16 + i + 8 * half;
        Y[(size_t)row * CLP + col] = acc[i];
    }
}

// Z[row][j*D+l] = sum_c Gc_j[c,d] * src_j[bn][c*D+l]; identity branches are copies.
// j = 2*gn + gc, gn in {I,Gs,Gs2}, gc in {I,Gc}; pads k in [6D,Kpad) with zeros.
__global__ void k_zbuild(const float* __restrict__ XH, const float* __restrict__ XN,
                         const float* __restrict__ Gc, float* __restrict__ Z,
                         int D, int Kpad, int CLp) {
    int idx = blockIdx.x * blockDim.x + threadIdx.x;
    if (idx >= ROWS_ * Kpad) return;
    int k = idx % Kpad, row = idx / Kpad;
    float v = 0.f;
    if (k < 6 * D) {
        int j = k / D, l = k % D;
        int gn = j >> 1, gc = j & 1;
        const float* src = (gn == 0) ? XH : XN + (size_t)(gn - 1) * BN_ * CLp;
        int d = row % C_;
        int bn = row / C_;
        const float* srow = src + (size_t)bn * CLp;
        if (gc == 0) {
            v = srow[d * D + l];
        } else {
            float s = 0.f;
            for (int c = 0; c < C_; ++c)
                s += Gc[c * C_ + d] * srow[c * D + l];
            v = s;
        }
    }
    Z[(size_t)row * Kpad + k] = v;
}

// Out[row][h] = Z[row][:] @ Wp[:, h] + bias[h]; Wp pre-padded -> guard-free
template<int KPAD>
__global__ __launch_bounds__(32) void k_gemm_zw(const float* __restrict__ Z,
                                                const float* __restrict__ Wp,
                                                const float* __restrict__ bias, int Nw,
                                                float* __restrict__ Out) {
    int lane = threadIdx.x, r = lane & 15, half = lane >> 4;
    int ctile = blockIdx.x, mtile = blockIdx.y;
    int arow = mtile * 16 + r;
    int col  = ctile * 16 + r;
    v8f acc = {0.f, 0.f, 0.f, 0.f, 0.f, 0.f, 0.f, 0.f};
    #pragma unroll 4
    for (int k4 = 0; k4 < KPAD; k4 += 4) {
        int k = k4 + 2 * half;
        v2f a = ld2(Z + (size_t)arow * KPAD + k);
        v2f b;
        b.x = Wp[(size_t)k * Nw + col];
        b.y = Wp[(size_t)(k + 1) * Nw + col];
        acc = wmma4(a, b, acc);
    }
    float bc = bias[col];
    for (int i = 0; i < 8; ++i) {
        int row = mtile * 16 + i + 8 * half;
        Out[(size_t)row * Nw + col] = acc[i] + bc;
    }
}

// update/reset gates; builds XH2 = [Xt | reset*H] in padded XH layout
__global__ void k_gates(const float* __restrict__ Gg, const float* __restrict__ H,
                        const float* __restrict__ Xseq, int t, int enc,
                        float* __restrict__ XH2, float* __restrict__ U,
                        int D, int Dx, int CLp) {
    int idx = blockIdx.x * blockDim.x + threadIdx.x;
    if (idx >= ROWS_ * HID_) return;
    int h = idx & (HID_ - 1), row = idx >> 5;
    int bn = row / C_, c = row % C_;
    float u  = sigf(Gg[(size_t)row * (2 * HID_) + h]);
    float rs = sigf(Gg[(size_t)row * (2 * HID_) + HID_ + h]);
    U[idx] = u;
    size_t base = (size_t)bn * CLp + c * D;
    XH2[base + Dx + h] = rs * H[idx];
    if (enc) {
        if (h == 0) {
            int b = bn / N_, n = bn % N_;
            XH2[base] = Xseq[(size_t)((b * T_ + t) * N_ + n) * C_ + c];
        }
    } else {
        XH2[base + h] = H[idx];   // decoder: Xt == H, Dx == 32
    }
}

// H = (1-u)*H + u*tanh(cand_pre)
__global__ void k_update(float* __restrict__ H, const float* __restrict__ U,
                         const float* __restrict__ CP) {
    int idx = blockIdx.x * blockDim.x + threadIdx.x;
    if (idx >= ROWS_ * HID_) return;
    float u = U[idx];
    H[idx] = (1.f - u) * H[idx] + u * tanhf(CP[idx]);
}

// out[b,step,n,c] = (H[row] @ W1 + b1) @ W2 + b2
__global__ void k_dec_out(const float* __restrict__ H, const float* __restrict__ W1,
                          const float* __restrict__ b1, const float* __restrict__ W2,
                          const float* __restrict__ b2, float* __restrict__ out, int step) {
    int row = blockIdx.x * blockDim.x + threadIdx.x;
    if (row >= ROWS_) return;
    const float* h = H + (size_t)row * HID_;
    float o = b2[0];
    for (int k = 0; k < 16; ++k) {
        float s = b1[k];
        for (int i = 0; i < HID_; ++i) s += h[i] * W1[i * 16 + k];
        o += s * W2[k];
    }
    int b = row / (N_ * C_), nc = row % (N_ * C_);
    out[(size_t)(b * OUTH_ + step) * (N_ * C_) + nc] = o;
}

// ---------------- workspace layout (floats, all offsets even) ----------------
#define OFF_US   0u
#define OFF_VS   (OFF_US   + (unsigned)(BT_*N_*HID_))        // 245760
#define OFF_UC   (OFF_VS   + (unsigned)(BT_*N_*HID_))
#define OFF_VC   (OFF_UC   + (unsigned)(BT_*16*HID_))        // 49152
#define OFF_MS   (OFF_VC   + (unsigned)(BT_*16*HID_))
#define OFF_PS   (OFF_MS   + (unsigned)N2_)
#define OFF_GS   (OFF_PS   + (unsigned)N2_)
#define OFF_GS2  (OFF_GS   + (unsigned)N2_)
#define OFF_MC   (OFF_GS2  + (unsigned)N2_)
#define OFF_PC   (OFF_MC   + (unsigned)C2_)
#define OFF_GC   (OFF_PC   + (unsigned)C2_)
#define OFF_PART (OFF_GC   + (unsigned)C2_)
#define OFF_WGE  (OFF_PART + (unsigned)(ICH_*N2_))           // 256000
#define OFF_WCE  (OFF_WGE  + (unsigned)(200*64))
#define OFF_WGD  (OFF_WCE  + (unsigned)(200*32))
#define OFF_WCD  (OFF_WGD  + (unsigned)(384*64))
#define OFF_H    (OFF_WCD  + (unsigned)(384*32))
#define OFF_UG   (OFF_H    + (unsigned)(ROWS_*HID_))
#define OFF_XH   (OFF_UG   + (unsigned)(ROWS_*HID_))
#define OFF_XN   (OFF_XH   + (unsigned)(BN_*768))
#define OFF_Z    (OFF_XN   + (unsigned)(2*BN_*768))
#define OFF_GB   (OFF_Z    + (unsigned)(ROWS_*384))
#define OFF_CP   (OFF_GB   + (unsigned)(ROWS_*64))

// one GRU cell (templated on compile-time extents -> unrolled, guard-free WMMA)
template<int D, int DX, int CLP, int KPAD>
static void launch_cell(hipStream_t stream, const float* X, int t, int enc,
                        float* pH, float* pXH, float* pXN, float* pZ, float* pGB,
                        float* pCP, float* pUG, const float* pGS, const float* pGS2,
                        const float* pGC, const float* Wgp, const float* bg,
                        const float* Wcp, const float* bc) {
    constexpr int CT = CLP / 16;
    if (enc) k_xh_enc<<<(BN_*CLP + 255)/256, 256, 0, stream>>>(X, t, pH, pXH);
    else     k_xh_dec<<<(BN_*CLP + 255)/256, 256, 0, stream>>>(pH, pXH);
    // gates
    k_diffuse<CLP><<<dim3(CT, 5, 2*B_), 32, 0, stream>>>(pXH, pGS, pGS2, pXN);
    k_zbuild <<<(ROWS_*KPAD + 255)/256, 256, 0, stream>>>(pXH, pXN, pGC, pZ, D, KPAD, CLP);
    k_gemm_zw<KPAD><<<dim3(4, ROWS_/16), 32, 0, stream>>>(pZ, Wgp, bg, 2*HID_, pGB);
    k_gates  <<<(ROWS_*HID_ + 255)/256, 256, 0, stream>>>(pGB, pH, X, t, enc, pXH, pUG,
                                                          D, DX, CLP);
    // candidate
    k_diffuse<CLP><<<dim3(CT, 5, 2*B_), 32, 0, stream>>>(pXH, pGS, pGS2, pXN);
    k_zbuild <<<(ROWS_*KPAD + 255)/256, 256, 0, stream>>>(pXH, pXN, pGC, pZ, D, KPAD, CLP);
    k_gemm_zw<KPAD><<<dim3(2, ROWS_/16), 32, 0, stream>>>(pZ, Wcp, bc, HID_, pCP);
    k_update <<<(ROWS_*HID_ + 255)/256, 256, 0, stream>>>(pH, pUG, pCP);
}

extern "C" void kernel_launch(void* const* d_in, const int* in_sizes, int n_in,
                              void* d_out, int out_size, void* d_ws, size_t ws_size,
                              hipStream_t stream) {
    const float* X    = (const float*)d_in[0];
    const float* As   = (const float*)d_in[1];
    const float* Ac   = (const float*)d_in[2];
    const float* WuS  = (const float*)d_in[3];
    const float* WvS  = (const float*)d_in[4];
    const float* WuC  = (const float*)d_in[5];
    const float* WvC  = (const float*)d_in[6];
    const float* WA_S = (const float*)d_in[7];
    const float* bA_S = (const float*)d_in[8];
    const float* WP_S = (const float*)d_in[9];
    const float* bP_S = (const float*)d_in[10];
    const float* WA_C = (const float*)d_in[11];
    const float* bA_C = (const float*)d_in[12];
    const float* WP_C = (const float*)d_in[13];
    const float* bP_C = (const float*)d_in[14];
    const float* eWg  = (const float*)d_in[15];
    const float* ebg  = (const float*)d_in[16];
    const float* eWc  = (const float*)d_in[17];
    const float* ebc  = (const float*)d_in[18];
    const float* dWg  = (const float*)d_in[19];
    const float* dbg  = (const float*)d_in[20];
    const float* dWc  = (const float*)d_in[21];
    const float* dbc  = (const float*)d_in[22];
    const float* W1   = (const float*)d_in[23];
    const float* b1   = (const float*)d_in[24];
    const float* W2   = (const float*)d_in[25];
    const float* b2   = (const float*)d_in[26];
    float* out = (float*)d_out;
    float* ws  = (float*)d_ws;

    float* pUS = ws + OFF_US;  float* pVS = ws + OFF_VS;
    float* pUC = ws + OFF_UC;  float* pVC = ws + OFF_VC;
    float* pMS = ws + OFF_MS;  float* pPS = ws + OFF_PS;
    float* pGS = ws + OFF_GS;  float* pGS2= ws + OFF_GS2;
    float* pMC = ws + OFF_MC;  float* pPC = ws + OFF_PC;  float* pGC = ws + OFF_GC;
    float* pPART = ws + OFF_PART;
    float* pWGE = ws + OFF_WGE; float* pWCE = ws + OFF_WCE;
    float* pWGD = ws + OFF_WGD; float* pWCD = ws + OFF_WCD;
    float* pH  = ws + OFF_H;   float* pUG = ws + OFF_UG;
    float* pXH = ws + OFF_XH;  float* pXN = ws + OFF_XN;
    float* pZ  = ws + OFF_Z;   float* pGB = ws + OFF_GB;  float* pCP = ws + OFF_CP;

    // --- phase A: adaptive graphs ---
    k_uv_spatial<<<(BT_*N_*HID_ + 255)/256, 256, 0, stream>>>(X, WuS, WvS, pUS, pVS);
    k_uv_channel<<<(BT_*16*HID_ + 255)/256, 256, 0, stream>>>(X, WuC, WvC, pUC, pVC);
    k_mm_wmma<N_, N_><<<dim3(5,5), 32, 0, stream>>>(pUS, pVS, pMS);
    k_mm_wmma<C_, 16><<<dim3(1,1), 32, 0, stream>>>(pUC, pVC, pMC);
    k_softmax<<<N_, 128, 0, stream>>>(pMS, pPS, N_);
    k_softmax<<<C_, 128, 0, stream>>>(pMC, pPC, C_);
    k_fuse_partial<<<dim3(25, ICH_), 256, 0, stream>>>(As, WA_S, pPS, WP_S, pPART, N2_, N2_/ICH_);
    k_fuse_final  <<<25, 256, 0, stream>>>(pPART, ICH_, bA_S, bP_S, As, pPS, pGS, N2_);
    k_fuse_partial<<<dim3(1, 1), 256, 0, stream>>>(Ac, WA_C, pPC, WP_C, pPART, C2_, C2_);
    k_fuse_final  <<<1, 256, 0, stream>>>(pPART, 1, bA_C, bP_C, Ac, pPC, pGC, C2_);
    k_cheby<<<dim3(5,5), 32, 0, stream>>>(pGS, pGS2);
    k_fill0<<<(ROWS_*HID_ + 255)/256, 256, 0, stream>>>(pH, ROWS_*HID_);
    // pad GRU weights once -> guard-free GEMM loads
    k_padw<<<(200*64 + 255)/256, 256, 0, stream>>>(eWg, 198, 64, pWGE, 200);
    k_padw<<<(200*32 + 255)/256, 256, 0, stream>>>(eWc, 198, 32, pWCE, 200);
    k_padw<<<(384*64 + 255)/256, 256, 0, stream>>>(dWg, 384, 64, pWGD, 384);
    k_padw<<<(384*32 + 255)/256, 256, 0, stream>>>(dWc, 384, 32, pWCD, 384);

    // --- phase B: GRU encoder (12 steps) + decoder (6 steps) ---
    for (int t = 0; t < T_; ++t)
        launch_cell<33, 1, 400, 200>(stream, X, t, 1, pH, pXH, pXN, pZ, pGB, pCP, pUG,
                                     pGS, pGS2, pGC, pWGE, ebg, pWCE, ebc);
    for (int s = 0; s < OUTH_; ++s) {
        launch_cell<64, 32, 768, 384>(stream, X, 0, 0, pH, pXH, pXN, pZ, pGB, pCP, pUG,
                                      pGS, pGS2, pGC, pWGD, dbg, pWCD, dbc);
        k_dec_out<<<(ROWS_ + 255)/256, 256, 0, stream>>>(pH, W1, b1, W2, b2, out, s);
    }
    (void)in_sizes; (void)n_in; (void)out_size; (void)ws_size;
}